// ROIPoolingLayer_523986010556
// MI455X (gfx1250) — compile-verified
//
#include <hip/hip_runtime.h>

// ROI max-pooling, MI455X (gfx1250).
// fm (4,64,64,256) f32, rois (4,128,4) f32 -> out (4,128,7,7,256) f32.
//
// Pure gather/max-reduce: no matmul structure -> WMMA inapplicable; reuse
// factor of fm elements is exactly 1 per ROI (bins partition the region), so
// async-to-LDS staging cannot reduce L2 traffic and would only add overhead.
// Optimal path: direct fully-coalesced global_load_b128, maximum loads in
// flight, fine-grained blocks for load balance.
//
// Grid: one block per (ROI, ph-band) = 4*128*7 = 3584 blocks.
// Block: 448 threads = 7 pw-bins x 64 channel-quads (14 wave32); each wave32
// reads one contiguous 512B line per (y,x) cell.

#define BB 4
#define RR 128
#define HH 64
#define WW 64
#define CC 256
#define PH 7
#define PW 7
#define CQ (CC / 4)        // 64 channel quads
#define BLOCK (PW * CQ)    // 448 threads = 14 wave32

__device__ __forceinline__ void vmax4(float4& a, const float4 v) {
    a.x = fmaxf(a.x, v.x);
    a.y = fmaxf(a.y, v.y);
    a.z = fmaxf(a.z, v.z);
    a.w = fmaxf(a.w, v.w);
}

__global__ __launch_bounds__(BLOCK) void roipool_max_kernel(
    const float* __restrict__ fm,
    const float* __restrict__ rois,
    float* __restrict__ out)
{
    const int blk = blockIdx.x;        // (b*RR + r)*PH + ph
    const int ph  = blk % PH;
    const int ri  = blk / PH;          // b*RR + r
    const int b   = ri >> 7;           // RR == 128
    const int tid = threadIdx.x;
    const int pw  = tid / CQ;          // 0..6
    const int cq  = tid % CQ;          // 0..63

    // ROI coords (block-uniform -> scalar loads)
    const float* roi = rois + (size_t)ri * 4;
    const float y0 = roi[0], x0 = roi[1], y1 = roi[2], x1 = roi[3];
    const int hs = (int)(HH * y0);
    const int ws = (int)(WW * x0);
    const int he = (int)(HH * y1);
    const int we = (int)(WW * x1);

    // step = floor((e-s)/7); degenerate region -> step 0 -> bins 0..5 empty,
    // bin 6 covers [s,e)  (matches reference seg-id logic; empty bin => -inf,
    // matching jax.ops.segment_max fill)
    const int sh = (he > hs) ? (he - hs) / PH : 0;
    const int sw = (we > ws) ? (we - ws) / PW : 0;

    const int rs = hs + ph * sh;
    int re = (ph == PH - 1) ? he : (rs + sh);
    if (re > he) re = he;

    const int cs = ws + pw * sw;
    int ce = (pw == PW - 1) ? we : (cs + sw);
    if (ce > we) ce = we;

    const float* fmb = fm + (size_t)b * (HH * WW * CC) + (size_t)cq * 4;

    // gfx1250 prefetch of this band's first line toward WGP caches
    if (rs < re && cs < ce) {
        __builtin_prefetch(fmb + (size_t)rs * (WW * CC) + (size_t)cs * CC, 0, 0);
    }

    float4 a0 = make_float4(-INFINITY, -INFINITY, -INFINITY, -INFINITY);
    float4 a1 = a0;

    for (int y = rs; y < re; ++y) {
        const float* p = fmb + (size_t)y * (WW * CC) + (size_t)cs * CC;
        int x = cs;
        // two independent accumulator chains -> 2x loads in flight per lane
        for (; x + 1 < ce; x += 2) {
            const float4 v0 = *(const float4*)(p);
            const float4 v1 = *(const float4*)(p + CC);
            vmax4(a0, v0);
            vmax4(a1, v1);
            p += 2 * CC;
        }
        if (x < ce) {
            vmax4(a0, *(const float4*)(p));
        }
    }
    vmax4(a0, a1);

    float* outp = out + (((size_t)blk) * PW + pw) * CC + (size_t)cq * 4;
    *(float4*)(outp) = a0;             // coalesced b128 store
}

extern "C" void kernel_launch(void* const* d_in, const int* in_sizes, int n_in,
                              void* d_out, int out_size, void* d_ws, size_t ws_size,
                              hipStream_t stream) {
    (void)in_sizes; (void)n_in; (void)d_ws; (void)ws_size; (void)out_size;
    const float* fm   = (const float*)d_in[0];
    const float* rois = (const float*)d_in[1];
    float*       out  = (float*)d_out;

    dim3 grid(BB * RR * PH);   // 3584 workgroups, one per (ROI, ph-band)
    dim3 block(BLOCK);         // 448 threads (14 wave32)
    roipool_max_kernel<<<grid, block, 0, stream>>>(fm, rois, out);
}